// TotalPairwiseConfusion_77936476553589
// MI455X (gfx1250) — compile-verified
//
#include <hip/hip_runtime.h>

typedef float v2f __attribute__((ext_vector_type(2)));
typedef float v4f __attribute__((ext_vector_type(4)));
typedef float v8f __attribute__((ext_vector_type(8)));

#define DIMK   1024
#define NROWS  8192
#define HALFN  4096
#define NT32   128   // 4096 / 32 : 32x32 wave tiles per output dimension

// ---------------------------------------------------------------------------
// Pass 1: per-row inverse norm and normalized squared-norm.
//   inv[r] = 1 / max(sqrt(sum f^2), 1e-12)
//   sq[r]  = sum(f^2) * inv[r]^2            ( == |normalized row|^2 )
// ---------------------------------------------------------------------------
__global__ void TPC_rownorm_kernel(const float* __restrict__ feat,
                                   float* __restrict__ inv,
                                   float* __restrict__ sq) {
  __shared__ float red[256];
  const int row = blockIdx.x;
  const float* p = feat + (size_t)row * DIMK;
  float s = 0.f;
  for (int k = threadIdx.x; k < DIMK; k += 256) {
    float x = p[k];
    s += x * x;
  }
  red[threadIdx.x] = s;
  __syncthreads();
  for (int off = 128; off > 0; off >>= 1) {
    if (threadIdx.x < off) red[threadIdx.x] += red[threadIdx.x + off];
    __syncthreads();
  }
  if (threadIdx.x == 0) {
    float n2    = red[0];
    float denom = fmaxf(sqrtf(n2), 1e-12f);
    float iv    = 1.0f / denom;
    inv[row] = iv;
    sq[row]  = n2 * iv * iv;
  }
}

#define WMMA_F32(cacc, av, bv) \
  (cacc) = __builtin_amdgcn_wmma_f32_16x16x4_f32(false, (av), false, (bv), \
                                                 (short)0, (cacc), false, false)

// ---------------------------------------------------------------------------
// Pass 2: one wave32 per 32x32 tile (2x2 register tile of 16x16 WMMA tiles)
// of the 4096x4096 cross-Gram matrix. V_WMMA_F32_16X16X4_F32, K swept in
// steps of 8. Per k-step: 4 x global_load_b128 -> 8 WMMAs (1 load : 2 wmma).
// The K axis may be permuted freely as long as A and B agree, so each lane
// loads a contiguous float4 at k + 4*(lane>>4) and feeds (x,y) to one WMMA
// and (z,w) to the next.
// ---------------------------------------------------------------------------
__global__ void TPC_pairwise_kernel(const float* __restrict__ feat,
                                    const float* __restrict__ inv,
                                    const float* __restrict__ sq,
                                    double* __restrict__ partials) {
  const int lane = threadIdx.x & 31;
  const int wid  = threadIdx.x >> 5;
  const int gw   = blockIdx.x * 8 + wid;   // global 32x32-tile id, 0..16383
  const int ti   = gw / NT32;              // left-row 32-tile
  const int tj   = gw % NT32;              // right-row 32-tile
  const int r15  = lane & 15;
  const int s    = lane >> 4;

  const float* L0 = feat + (size_t)(ti * 32 + r15) * DIMK + s * 4;
  const float* L1 = L0 + (size_t)16 * DIMK;
  const float* R0 = feat + (size_t)(HALFN + tj * 32 + r15) * DIMK + s * 4;
  const float* R1 = R0 + (size_t)16 * DIMK;

  v8f c00 = {}, c01 = {}, c10 = {}, c11 = {};
#pragma unroll 2
  for (int k = 0; k < DIMK; k += 8) {
    v4f a0 = *(const v4f*)(L0 + k);
    v4f a1 = *(const v4f*)(L1 + k);
    v4f b0 = *(const v4f*)(R0 + k);
    v4f b1 = *(const v4f*)(R1 + k);
    v2f a0lo = {a0.x, a0.y}, a0hi = {a0.z, a0.w};
    v2f a1lo = {a1.x, a1.y}, a1hi = {a1.z, a1.w};
    v2f b0lo = {b0.x, b0.y}, b0hi = {b0.z, b0.w};
    v2f b1lo = {b1.x, b1.y}, b1hi = {b1.z, b1.w};
    // 4 independent accumulator chains interleaved to hide WMMA latency.
    WMMA_F32(c00, a0lo, b0lo);
    WMMA_F32(c01, a0lo, b1lo);
    WMMA_F32(c10, a1lo, b0lo);
    WMMA_F32(c11, a1lo, b1lo);
    WMMA_F32(c00, a0hi, b0hi);
    WMMA_F32(c01, a0hi, b1hi);
    WMMA_F32(c10, a1hi, b0hi);
    WMMA_F32(c11, a1hi, b1hi);
  }

  // Epilogue: C/D layout -> VGPR r, lane L holds (M = r + 8*(L>>4), N = L&15).
  const int j0 = HALFN + tj * 32 + r15;
  const float invR0 = inv[j0],      sqR0 = sq[j0];
  const float invR1 = inv[j0 + 16], sqR1 = sq[j0 + 16];
  float local = 0.f;
#pragma unroll
  for (int r = 0; r < 8; ++r) {
    const int i0 = ti * 32 + r + s * 8;
    const int i1 = i0 + 16;
    const float invL0 = inv[i0], sqL0 = sq[i0];
    const float invL1 = inv[i1], sqL1 = sq[i1];
    local += fmaxf(sqL0 + sqR0 - 2.0f * c00[r] * invL0 * invR0, 0.0f);
    local += fmaxf(sqL0 + sqR1 - 2.0f * c01[r] * invL0 * invR1, 0.0f);
    local += fmaxf(sqL1 + sqR0 - 2.0f * c10[r] * invL1 * invR0, 0.0f);
    local += fmaxf(sqL1 + sqR1 - 2.0f * c11[r] * invL1 * invR1, 0.0f);
  }

  // wave32 reduction
  for (int off = 16; off > 0; off >>= 1)
    local += __shfl_xor(local, off, 32);

  __shared__ double wpart[8];
  if (lane == 0) wpart[wid] = (double)local;
  __syncthreads();
  if (threadIdx.x == 0) {
    double t = 0.0;
#pragma unroll
    for (int w = 0; w < 8; ++w) t += wpart[w];
    partials[blockIdx.x] = t;   // deterministic: no atomics
  }
}

// ---------------------------------------------------------------------------
// Pass 3: deterministic final reduction of per-block doubles -> f32 scalar.
// ---------------------------------------------------------------------------
__global__ void TPC_reduce_kernel(const double* __restrict__ partials,
                                  float* __restrict__ out, int n) {
  __shared__ double red[256];
  double s = 0.0;
  for (int i = threadIdx.x; i < n; i += 256) s += partials[i];
  red[threadIdx.x] = s;
  __syncthreads();
  for (int off = 128; off > 0; off >>= 1) {
    if (threadIdx.x < off) red[threadIdx.x] += red[threadIdx.x + off];
    __syncthreads();
  }
  if (threadIdx.x == 0) out[0] = (float)red[0];
}

// ---------------------------------------------------------------------------
extern "C" void kernel_launch(void* const* d_in, const int* in_sizes, int n_in,
                              void* d_out, int out_size, void* d_ws, size_t ws_size,
                              hipStream_t stream) {
  const float* feat = (const float*)d_in[0];
  float* out = (float*)d_out;

  char* ws = (char*)d_ws;
  float*  inv      = (float*)(ws);                 // 8192 f32  (32 KB)
  float*  sq       = (float*)(ws + 32768);         // 8192 f32  (32 KB)
  double* partials = (double*)(ws + 65536);        // 2048 f64  (16 KB)

  const int nblocks = (NT32 * NT32) / 8;           // 2048 blocks x 8 waves
  TPC_rownorm_kernel<<<NROWS, 256, 0, stream>>>(feat, inv, sq);
  TPC_pairwise_kernel<<<nblocks, 256, 0, stream>>>(feat, inv, sq, partials);
  TPC_reduce_kernel<<<1, 256, 0, stream>>>(partials, out, nblocks);
}